// Local2DAttentionLayer_31911607009651
// MI455X (gfx1250) — compile-verified
//
#include <hip/hip_runtime.h>
#include <hip/hip_bf16.h>
#include <math.h>

typedef __attribute__((ext_vector_type(2))) float v2f;
typedef __attribute__((ext_vector_type(8))) float v8f;

#define E_   192
#define H_   64
#define W_   64
#define L_   (H_ * W_)          // 4096
#define NB_  2
#define NH_  6
#define DK_  32
#define KS_  7
#define P_   49

// ---------------------------------------------------------------------------
// Stage 1: q/k/v = W * x + b  as WMMA f32 16x16x4 GEMM.
//   x   : [N, E, L]  (L = H*W), column-contiguous in l
//   Wm  : [E, E] row-major
//   out : qkv workspace [3][N][E][L]
// One wave computes a 16(M=o) x 64(columns l) tile; K-loop of 48 steps of 4.
// ---------------------------------------------------------------------------
__global__ __launch_bounds__(256) void qkv_gemm_kernel(
    const float* __restrict__ x,
    const float* __restrict__ Wq, const float* __restrict__ bq,
    const float* __restrict__ Wk, const float* __restrict__ bk,
    const float* __restrict__ Wv, const float* __restrict__ bv,
    float* __restrict__ qkv)
{
    const int lane = threadIdx.x & 31;
    const int wave = blockIdx.x * (blockDim.x >> 5) + (threadIdx.x >> 5);

    const int CT = L_ / 64;   // 64 column chunks of 64
    const int MT = E_ / 16;   // 12 row tiles
    int t = wave;
    const int ct  = t % CT; t /= CT;
    const int mt  = t % MT; t /= MT;
    const int n   = t % NB_; t /= NB_;
    const int mat = t;                     // 0=q 1=k 2=v
    if (mat >= 3) return;

    const float* Wm = (mat == 0) ? Wq : (mat == 1) ? Wk : Wv;
    const float* bm = (mat == 0) ? bq : (mat == 1) ? bk : bv;
    float* out = qkv + ((size_t)mat * NB_ + n) * E_ * L_;

    const int o0   = mt * 16;
    const int l0   = ct * 64;
    const int m    = lane & 15;    // M (for A) / N (for B) within tile
    const int kp   = lane >> 4;    // K-pair selector (0 -> K{0,1}, 1 -> K{2,3})

    const float* xn = x + (size_t)n * E_ * L_;

    v8f acc0 = {}, acc1 = {}, acc2 = {}, acc3 = {};
    for (int c0 = 0; c0 < E_; c0 += 4) {
        // A[m][k]: lane = m + 16*(k/2), vgpr = k%2  -> contiguous float2 of W row
        const float* wp = Wm + (size_t)(o0 + m) * E_ + (c0 + 2 * kp);
        v2f a; a.x = wp[0]; a.y = wp[1];
        // B[k][nn]: lane = nn + 16*(k/2), vgpr = k%2 -> two stride-L column loads
        const float* xp = xn + (size_t)(c0 + 2 * kp) * L_ + l0 + m;
        v2f b0; b0.x = xp[0];  b0.y = xp[L_ + 0];
        v2f b1; b1.x = xp[16]; b1.y = xp[L_ + 16];
        v2f b2; b2.x = xp[32]; b2.y = xp[L_ + 32];
        v2f b3; b3.x = xp[48]; b3.y = xp[L_ + 48];
        acc0 = __builtin_amdgcn_wmma_f32_16x16x4_f32(false, a, false, b0, (short)0, acc0, false, false);
        acc1 = __builtin_amdgcn_wmma_f32_16x16x4_f32(false, a, false, b1, (short)0, acc1, false, false);
        acc2 = __builtin_amdgcn_wmma_f32_16x16x4_f32(false, a, false, b2, (short)0, acc2, false, false);
        acc3 = __builtin_amdgcn_wmma_f32_16x16x4_f32(false, a, false, b3, (short)0, acc3, false, false);
    }

    // D element (vgpr p, lane): row = o0 + p + 8*(lane>>4), col = l0 + 16*t + (lane&15)
    const int half = lane >> 4;
#pragma unroll
    for (int p = 0; p < 8; ++p) {
        const int   row  = o0 + p + 8 * half;
        const float bias = bm[row];
        float* op = out + (size_t)row * L_ + l0 + m;
        op[0]  = acc0[p] + bias;
        op[16] = acc1[p] + bias;
        op[32] = acc2[p] + bias;
        op[48] = acc3[p] + bias;
    }
}

// ---------------------------------------------------------------------------
// Stage 2: neighborhood attention with online softmax.
// One thread per (n, head, pixel). Lanes = consecutive x -> coalesced column
// loads of k/v (stride-L rows contiguous across wave).
// Zero-padded (out-of-image) neighbors contribute score = rpb (dot == 0) and
// V == 0, matching the reference's unfold-with-zero-padding semantics.
// ---------------------------------------------------------------------------
__global__ __launch_bounds__(256) void local_attn_kernel(
    const float* __restrict__ qbuf,  // [N, E, L]
    const float* __restrict__ kbuf,  // [N, E, L]
    const float* __restrict__ vbuf,  // [N, E, L]
    const float* __restrict__ rpb,   // [Nh, 1, P]
    float* __restrict__ out,         // [N, E, L]
    float* __restrict__ scores)      // [N, L, Nh, 1, P]
{
    const int t = blockIdx.x * blockDim.x + threadIdx.x;
    if (t >= NB_ * NH_ * L_) return;
    const int l = t % L_;
    const int h = (t / L_) % NH_;
    const int n = t / (L_ * NH_);
    const int y  = l / W_;
    const int xw = l % W_;

    const float scale = 0.07216878364870323f;  // 1/sqrt(192)

    const size_t hb   = ((size_t)n * E_ + h * DK_) * L_;  // head base into [N,E,L]
    const size_t base = hb + l;

    float q[DK_];
#pragma unroll
    for (int d = 0; d < DK_; ++d) q[d] = qbuf[base + (size_t)d * L_];

    float mrun = -INFINITY, se = 0.0f;
    float acc[DK_];
#pragma unroll
    for (int d = 0; d < DK_; ++d) acc[d] = 0.0f;

    float* sc = scores + ((size_t)(n * L_ + l) * NH_ + h) * P_;
    const float* rp = rpb + h * P_;

    for (int p = 0; p < P_; ++p) {
        const int yy = y  + (p / KS_) - (KS_ / 2);
        const int xx = xw + (p % KS_) - (KS_ / 2);
        const bool ib = (yy >= 0) & (yy < H_) & (xx >= 0) & (xx < W_);

        float s = rp[p];
        if (ib) {
            const float* kp2 = kbuf + hb + (size_t)(yy * W_ + xx);
            float dot = 0.0f;
#pragma unroll
            for (int d = 0; d < DK_; ++d)
                dot = fmaf(q[d], kp2[(size_t)d * L_], dot);
            s += dot * scale;
        }
        sc[p] = s;

        const float mn   = fmaxf(mrun, s);
        const float corr = __expf(mrun - mn);   // exp(-inf)=0 on first iter
        const float w    = __expf(s - mn);
        se = se * corr + w;
        if (ib) {
            const float* vp = vbuf + hb + (size_t)(yy * W_ + xx);
#pragma unroll
            for (int d = 0; d < DK_; ++d)
                acc[d] = acc[d] * corr + w * vp[(size_t)d * L_];
        } else {
#pragma unroll
            for (int d = 0; d < DK_; ++d) acc[d] = acc[d] * corr;
        }
        mrun = mn;
    }

    const float inv = 1.0f / se;
#pragma unroll
    for (int d = 0; d < DK_; ++d)
        out[base + (size_t)d * L_] = acc[d] * inv;
}

extern "C" void kernel_launch(void* const* d_in, const int* in_sizes, int n_in,
                              void* d_out, int out_size, void* d_ws, size_t ws_size,
                              hipStream_t stream) {
    const float* x   = (const float*)d_in[0];
    const float* Wq  = (const float*)d_in[1];
    const float* bq  = (const float*)d_in[2];
    const float* Wk  = (const float*)d_in[3];
    const float* bk  = (const float*)d_in[4];
    const float* Wv  = (const float*)d_in[5];
    const float* bv  = (const float*)d_in[6];
    const float* rpb = (const float*)d_in[7];

    float* out    = (float*)d_out;                               // [N,E,H,W]
    float* scores = out + (size_t)NB_ * E_ * L_;                 // [N,L,Nh,1,P]
    float* qkv    = (float*)d_ws;                                // 3 * N*E*L floats (18 MB)

    // Stage 1: 3 matrices * 2 batches * 12 row-tiles * 64 col-chunks = 4608 waves
    const int waves  = 3 * NB_ * (E_ / 16) * (L_ / 64);
    const int blocks = waves / 8;  // 8 waves (256 threads) per block
    qkv_gemm_kernel<<<blocks, 256, 0, stream>>>(x, Wq, bq, Wk, bk, Wv, bv, qkv);

    const float* qb = qkv;
    const float* kb = qkv + (size_t)NB_ * E_ * L_;
    const float* vb = qkv + (size_t)2 * NB_ * E_ * L_;

    const int threads = NB_ * NH_ * L_;  // 49152
    local_attn_kernel<<<(threads + 255) / 256, 256, 0, stream>>>(qb, kb, vb, rpb, out, scores);
}